// CloudRasterizerOversample_49675591745756
// MI455X (gfx1250) — compile-verified
//
#include <hip/hip_runtime.h>
#include <hip/hip_bf16.h>

// Grid constants matching the reference
constexpr int   kNPixLo  = 256;
constexpr int   kNPixHi  = 1024;       // 256 * 4
constexpr int   kNv      = 64;
constexpr float kPixHi   = 0.025f;     // 0.1 / 4
constexpr float kFovHalf = 12.7875f;   // 0.5 * 1023 * 0.025
constexpr float kDv      = 10.0f;
constexpr float kVel0    = 0.0f;
constexpr int   kPlaneLo = kNPixLo * kNPixLo;   // 65536
// output size = kNv * kPlaneLo = 4194304 floats (16.8 MB -> L2 resident)

__global__ __launch_bounds__(256) void zero_out_kernel(float* __restrict__ out, int n) {
    int i  = blockIdx.x * blockDim.x + threadIdx.x;
    int i4 = i * 4;
    if (i4 + 3 < n) {
        *reinterpret_cast<float4*>(out + i4) = make_float4(0.f, 0.f, 0.f, 0.f);
    } else {
        for (int k = i4; k < n; ++k) out[k] = 0.f;
    }
}

__device__ __forceinline__ void fadd(float* p, float v) {
    // Native FP32 global atomic (global_atomic_add_f32), not a CAS loop.
    unsafeAtomicAdd(p, v);
}

__global__ __launch_bounds__(256) void cloud_scatter_kernel(
    const float* __restrict__ ra,  const float* __restrict__ dec,
    const float* __restrict__ vel, const float* __restrict__ flux,
    float* __restrict__ out, int M) {

    int i = blockIdx.x * blockDim.x + threadIdx.x;
    if (i >= M) return;

    // One speculative prefetch per wave, ~8 blocks ahead (gfx1250 global_prefetch_b8).
    if ((threadIdx.x & 31) == 0) {
        __builtin_prefetch(ra   + i + 2048, 0, 0);
        __builtin_prefetch(dec  + i + 2048, 0, 0);
        __builtin_prefetch(vel  + i + 2048, 0, 0);
        __builtin_prefetch(flux + i + 2048, 0, 0);
    }

    // Continuous hi-res grid coordinates (float32 math mirrors the reference).
    float x = (ra[i]  + kFovHalf) / kPixHi;
    float y = (dec[i] + kFovHalf) / kPixHi;
    float v = (vel[i] - kVel0)    / kDv;

    float xf = floorf(x), yf = floorf(y), vf = floorf(v);
    int ix0 = (int)xf, iy0 = (int)yf, iv0 = (int)vf;
    float fx = x - xf, fy = y - yf, fv = v - vf;

    // In-bounds mask: out-of-bounds points contribute nothing.
    if (ix0 < 0 || ix0 >= kNPixHi - 1 ||
        iy0 < 0 || iy0 >= kNPixHi - 1 ||
        iv0 < 0 || iv0 >= kNv - 1)
        return;

    // Fold the 4x4 box-mean into the scatter: weight *= 1/16, bin = hi>>2.
    float f = flux[i] * 0.0625f;

    int jx0 = ix0 >> 2, jx1 = (ix0 + 1) >> 2;
    int jy0 = iy0 >> 2, jy1 = (iy0 + 1) >> 2;
    bool dx = (jx1 != jx0);     // only when (ix0 & 3) == 3
    bool dy = (jy1 != jy0);

    // When both x-corners land in the same lo-res pixel, (1-fx)+fx == 1.
    float wxA = dx ? (1.0f - fx) : 1.0f;
    float wyA = dy ? (1.0f - fy) : 1.0f;

    float wv0 = 1.0f - fv;
    float wv1 = fv;

    float* out0 = out + iv0 * kPlaneLo;   // velocity plane iv0
    float* out1 = out0 + kPlaneLo;        // velocity plane iv0 + 1

    {
        int   p = jy0 * kNPixLo + jx0;
        float w = f * wxA * wyA;
        fadd(out0 + p, w * wv0);
        fadd(out1 + p, w * wv1);
    }
    if (dx) {
        int   p = jy0 * kNPixLo + jx1;
        float w = f * fx * wyA;
        fadd(out0 + p, w * wv0);
        fadd(out1 + p, w * wv1);
    }
    if (dy) {
        int   p = jy1 * kNPixLo + jx0;
        float w = f * wxA * fy;
        fadd(out0 + p, w * wv0);
        fadd(out1 + p, w * wv1);
    }
    if (dx && dy) {
        int   p = jy1 * kNPixLo + jx1;
        float w = f * fx * fy;
        fadd(out0 + p, w * wv0);
        fadd(out1 + p, w * wv1);
    }
}

extern "C" void kernel_launch(void* const* d_in, const int* in_sizes, int n_in,
                              void* d_out, int out_size, void* d_ws, size_t ws_size,
                              hipStream_t stream) {
    const float* ra   = (const float*)d_in[0];
    const float* dec  = (const float*)d_in[1];
    const float* vel  = (const float*)d_in[2];
    const float* flux = (const float*)d_in[3];
    float*       out  = (float*)d_out;

    int M = in_sizes[0];

    // Zero the 64x256x256 output (harness poisons it before timing).
    {
        int n_threads = (out_size + 3) / 4;
        int blocks    = (n_threads + 255) / 256;
        zero_out_kernel<<<blocks, 256, 0, stream>>>(out, out_size);
    }

    // Direct scatter into the lo-res cube (downsample folded in).
    {
        int blocks = (M + 255) / 256;
        cloud_scatter_kernel<<<blocks, 256, 0, stream>>>(ra, dec, vel, flux, out, M);
    }
}